// Resonator_23656679866672
// MI455X (gfx1250) — compile-verified
//
#include <hip/hip_runtime.h>
#include <cmath>

// ---------------- problem constants ----------------
#define SRATE   48000.0f
#define NSAMP   7864320        // total samples
#define LDEL    960            // delay line length (48000/50)
#define KFRM    8192           // NSAMP / LDEL
// ---------------- filter block-scan ----------------
#define FC      256            // samples per thread (serial)
#define FTPB    256            // threads per workgroup
#define FSPAN   (FC * FTPB)    // 65536 samples per workgroup
#define FNWG    (NSAMP / FSPAN) // 120 workgroups
// ---------------- comb block-scan ------------------
#define CK      128            // frames per chunk
#define NCHK    (KFRM / CK)    // 64 chunks

static_assert(KFRM * LDEL == NSAMP, "shape");
static_assert(FNWG * FSPAN == NSAMP, "filter partition");
static_assert(NCHK * CK == KFRM, "comb partition");

typedef __attribute__((ext_vector_type(2))) float v2f;
typedef __attribute__((ext_vector_type(8))) float v8f;

__device__ __forceinline__ float coefA(const float* prm, int i) {
    return expf(-6.28318530717958647692f * prm[i] / SRATE);
}

// ============================================================
// Pass A1: per-thread local 2-state IIR scan (zero init) +
// LDS Hillis-Steele scan of affine offsets (transition matrices
// are constant powers of M, closed-form). Emits per-thread
// inclusive offsets and one aggregate per workgroup.
// ============================================================
__global__ void filt_pass1(const float* __restrict__ x, const float* __restrict__ prm,
                           float* __restrict__ incl, float* __restrict__ wgagg) {
    __shared__ float su[FTPB];
    __shared__ float sv[FTPB];
    const int tid = threadIdx.x;
    const int wg  = blockIdx.x;
    const float a1 = coefA(prm, 2);           // lowpass at f_low (feeds highpass)
    const float a2 = coefA(prm, 3);           // lowpass at f_high
    const float b1 = 1.0f - a1, b2 = 1.0f - a2;
    const float g  = -b2 * a1;                // M[1][0]

    const int base = wg * FSPAN + tid * FC;
    float u = 0.0f, v = 0.0f;
    for (int n = 0; n < FC; ++n) {
        const float xv = x[base + n];
        u = a1 * u + b1 * xv;
        v = a2 * v + b2 * (xv - u);           // bandpass output state
    }
    su[tid] = u; sv[tid] = v;
    __syncthreads();

    // inclusive scan over offsets; matrix for distance d is M^(d*FC)
    float p1 = powf(a1, (float)FC);
    float p2 = powf(a2, (float)FC);
    for (int d = 1; d < FTPB; d <<= 1) {
        const float q = g * (p1 - p2) / (a1 - a2);
        float pu = 0.0f, pv = 0.0f;
        if (tid >= d) { pu = su[tid - d]; pv = sv[tid - d]; }
        __syncthreads();
        if (tid >= d) {
            su[tid] = p1 * pu + su[tid];
            sv[tid] = q * pu + p2 * pv + sv[tid];
        }
        __syncthreads();
        p1 *= p1; p2 *= p2;                   // square: M^(2d*FC)
    }

    const int idx = wg * FTPB + tid;
    incl[2 * idx]     = su[tid];
    incl[2 * idx + 1] = sv[tid];
    if (tid == FTPB - 1) { wgagg[2 * wg] = su[tid]; wgagg[2 * wg + 1] = sv[tid]; }
}

// ============================================================
// Pass A2: serial exclusive scan over the 120 workgroup
// aggregates (tiny). Matrix = M^FSPAN (closed form).
// ============================================================
__global__ void filt_pass2(const float* __restrict__ prm, const float* __restrict__ wgagg,
                           float* __restrict__ wgcar) {
    if (threadIdx.x != 0 || blockIdx.x != 0) return;
    const float a1 = coefA(prm, 2);
    const float a2 = coefA(prm, 3);
    const float g  = -(1.0f - a2) * a1;
    const float p1 = powf(a1, (float)FSPAN);  // underflows to 0 for these cutoffs: fine
    const float p2 = powf(a2, (float)FSPAN);
    const float q  = g * (p1 - p2) / (a1 - a2);
    float u = 0.0f, v = 0.0f;
    for (int w = 0; w < FNWG; ++w) {
        wgcar[2 * w] = u; wgcar[2 * w + 1] = v;
        const float nu = p1 * u + wgagg[2 * w];
        const float nv = q  * u + p2 * v + wgagg[2 * w + 1];
        u = nu; v = nv;
    }
}

// ============================================================
// Pass A3: inject carries (closed-form M^(tid*FC) on the wg
// carry + stored exclusive offsets), emit filt into d_out.
// ============================================================
__global__ void filt_pass3(const float* __restrict__ x, const float* __restrict__ prm,
                           const float* __restrict__ incl, const float* __restrict__ wgcar,
                           float* __restrict__ filt) {
    const int tid = threadIdx.x;
    const int wg  = blockIdx.x;
    const float a1 = coefA(prm, 2);
    const float a2 = coefA(prm, 3);
    const float b1 = 1.0f - a1, b2 = 1.0f - a2;
    const float g  = -b2 * a1;

    const float uw = wgcar[2 * wg], vw = wgcar[2 * wg + 1];
    const int idx = wg * FTPB + tid;
    float ue = 0.0f, ve = 0.0f;
    if (tid > 0) { ue = incl[2 * (idx - 1)]; ve = incl[2 * (idx - 1) + 1]; }
    const float p1 = powf(a1, (float)(tid * FC));
    const float p2 = powf(a2, (float)(tid * FC));
    const float q  = g * (p1 - p2) / (a1 - a2);
    float u = p1 * uw + ue;
    float v = q  * uw + p2 * vw + ve;

    const int base = wg * FSPAN + tid * FC;
    for (int n = 0; n < FC; ++n) {
        const float xv = x[base + n];
        u = a1 * u + b1 * xv;
        v = a2 * v + b2 * (xv - u);
        filt[base + n] = v;
    }
}

// ============================================================
// Pass B1: per (128-frame chunk, column) local comb aggregate,
// zero initial state. Lanes map to consecutive columns -> coalesced.
// ============================================================
__global__ void comb_pass1(const float* __restrict__ filt, const float* __restrict__ prm,
                           float* __restrict__ agg) {
    const int gid   = blockIdx.x * blockDim.x + threadIdx.x;   // 0..61439
    const int chunk = gid / LDEL;
    const int col   = gid - chunk * LDEL;
    const float fb  = prm[1];
    float z = 0.0f;
    const int base = chunk * CK * LDEL + col;
    for (int t = 0; t < CK; ++t) z = fb * (z + filt[base + t * LDEL]);
    agg[chunk * LDEL + col] = z;
}

// ============================================================
// Pass B2: per-column exclusive scan over the 64 chunks
// (carry factor fb^CK). 960 threads, coalesced.
// ============================================================
__global__ void comb_pass2(const float* __restrict__ prm, const float* __restrict__ agg,
                           float* __restrict__ car) {
    const int col = blockIdx.x * blockDim.x + threadIdx.x;
    if (col >= LDEL) return;
    const float fb  = prm[1];
    const float fbC = powf(fb, (float)CK);
    float z = 0.0f;
    for (int c = 0; c < NCHK; ++c) {
        car[c * LDEL + col] = z;
        z = fbC * z + agg[c * LDEL + col];
    }
}

// ============================================================
// Pass B3: final pass via WMMA. One wave owns a (16 frames x
// 16 cols) tile stack inside a 128-frame chunk. Local comb
// contribution = A (16x16 lower-triangular Toeplitz of fb
// powers) x F (16x16 filt tile) computed as 4 chained
// V_WMMA_F32_16X16X4_F32; carry rows added as fb^row * z0.
// Per-wave 16-frame carry hop via ds_bpermute of row 15.
// filt aliases the output buffer (read tile -> WMMA -> write
// same addresses; each tile exclusively owned by one wave).
// EXEC stays all-ones throughout (WMMA requirement).
// ============================================================
__global__ void comb_pass3(const float* __restrict__ input, float* fo,
                           const float* __restrict__ prm, const float* __restrict__ ccar) {
    const float fb   = prm[1];
    const float gain = prm[4];
    const int wid  = blockIdx.x * (blockDim.x >> 5) + (threadIdx.x >> 5); // 0..3839
    const int lane = threadIdx.x & 31;
    const int chunk = wid % NCHK;        // 0..63
    const int cg    = wid / NCHK;        // 0..59
    const int col0  = cg * 16;
    const int r  = lane & 15;            // matrix row / column index
    const int hi = lane >> 4;            // 0: lanes 0-15, 1: lanes 16-31

    // fb^row scaling per D-register (rows i and i+8 per half-wave)
    const float fb2 = fb * fb, fb4 = fb2 * fb2, fb8 = fb4 * fb4;
    float fbp[8];
    fbp[0] = hi ? fb8 : 1.0f;
#pragma unroll
    for (int i = 1; i < 8; ++i) fbp[i] = fbp[i - 1] * fb;

    // A operand (16x4 slices, ISA layout: M=lane&15; K = 4k + {0,2}/{1,3})
    v2f a[4];
#pragma unroll
    for (int k = 0; k < 4; ++k) {
        const int c0 = 4 * k + 2 * hi;
        a[k].x = (c0     < r) ? powf(fb, (float)(r - c0))     : 0.0f;
        a[k].y = (c0 + 1 < r) ? powf(fb, (float)(r - c0 - 1)) : 0.0f;
    }

    float z0 = ccar[chunk * LDEL + col0 + r];          // z at chunk start, per column
    const int tilebase = chunk * CK * LDEL + col0;

    for (int blk = 0; blk < CK / 16; ++blk) {
        const int base = tilebase + blk * 16 * LDEL;
        // B operand: 4x16 slices of the filt tile (N=lane&15, K rows split per half)
        v2f b[4];
#pragma unroll
        for (int k = 0; k < 4; ++k) {
            const int fr = 4 * k + 2 * hi;
            b[k].x = fo[base + fr * LDEL + r];
            b[k].y = fo[base + (fr + 1) * LDEL + r];
        }
        v8f c = {0.f, 0.f, 0.f, 0.f, 0.f, 0.f, 0.f, 0.f};
        c = __builtin_amdgcn_wmma_f32_16x16x4_f32(false, a[0], false, b[0], (short)0, c, false, false);
        c = __builtin_amdgcn_wmma_f32_16x16x4_f32(false, a[1], false, b[1], (short)0, c, false, false);
        c = __builtin_amdgcn_wmma_f32_16x16x4_f32(false, a[2], false, b[2], (short)0, c, false, false);
        c = __builtin_amdgcn_wmma_f32_16x16x4_f32(false, a[3], false, b[3], (short)0, c, false, false);

        float zlast = 0.0f;
#pragma unroll
        for (int i = 0; i < 8; ++i) {
            const int row = i + 8 * hi;                 // C/D layout: VGPR i -> rows i / i+8
            const float z = fbp[i] * z0 + c[i];
            const int off = base + row * LDEL + r;
            fo[off] = (input[off] + z) * gain;
            if (i == 7) zlast = z;                      // lanes 16-31 hold row 15
        }
        // carry to next 16-frame block: z[16] = fb * (z[15] + f[15])
        const int bidx = (16 + r) << 2;
        const float z15 = __int_as_float(__builtin_amdgcn_ds_bpermute(bidx, __float_as_int(zlast)));
        const float f15 = __int_as_float(__builtin_amdgcn_ds_bpermute(bidx, __float_as_int(b[3].y)));
        z0 = fb * (z15 + f15);
    }
}

// ============================================================
extern "C" void kernel_launch(void* const* d_in, const int* in_sizes, int n_in,
                              void* d_out, int out_size, void* d_ws, size_t ws_size,
                              hipStream_t stream) {
    (void)in_sizes; (void)n_in; (void)out_size; (void)ws_size;
    const float* input = (const float*)d_in[0];
    const float* delay = (const float*)d_in[1];
    const float* prm   = (const float*)d_in[2];
    float* out = (float*)d_out;          // doubles as 'filt' staging (same size)

    float* wsf   = (float*)d_ws;
    float* incl  = wsf;                          // 2*FNWG*FTPB   = 61440 floats
    float* wgagg = incl  + 2 * FNWG * FTPB;      // 2*FNWG        =   240
    float* wgcar = wgagg + 2 * FNWG;             // 2*FNWG        =   240
    float* cagg  = wgcar + 2 * FNWG;             // NCHK*LDEL     = 61440
    float* ccar  = cagg  + NCHK * LDEL;          // NCHK*LDEL     = 61440
    // total ws usage: 184,800 floats = ~740 KB

    filt_pass1<<<dim3(FNWG), dim3(FTPB), 0, stream>>>(delay, prm, incl, wgagg);
    filt_pass2<<<dim3(1),    dim3(32),   0, stream>>>(prm, wgagg, wgcar);
    filt_pass3<<<dim3(FNWG), dim3(FTPB), 0, stream>>>(delay, prm, incl, wgcar, out);
    comb_pass1<<<dim3(240),  dim3(256),  0, stream>>>(out, prm, cagg);     // 240*256 = NCHK*LDEL
    comb_pass2<<<dim3(30),   dim3(32),   0, stream>>>(prm, cagg, ccar);    // 960 columns
    comb_pass3<<<dim3(480),  dim3(256),  0, stream>>>(input, out, prm, ccar); // 3840 waves = 64*60 tiles
}